// DBPGAT_41059887350099
// MI455X (gfx1250) — compile-verified
//
#include <hip/hip_runtime.h>
#include <math.h>

typedef float v2f __attribute__((ext_vector_type(2)));
typedef float v8f __attribute__((ext_vector_type(8)));

#define HID 256
#define NH1 8
#define NC1 32
#define NC2 16
#define NEG_SLOPE 0.2f

// ---------------- helpers ----------------

// float atomic max via signed/unsigned integer atomics (valid for non-NaN floats)
__device__ __forceinline__ void atomic_max_f32(float* addr, float v) {
    if (v >= 0.0f)
        atomicMax((int*)addr, __float_as_int(v));
    else
        atomicMin((unsigned int*)addr, __float_as_uint(v));
}

__device__ __forceinline__ void edge_src_dst(const int* __restrict__ ei, int E, int e,
                                             int& s, int& d) {
    if (e < E) { s = ei[e]; d = ei[E + e]; }
    else       { s = e - E; d = e - E; }     // self loops appended after real edges
}

// ---------------- init accumulators ----------------

__global__ void dbp_init(float* __restrict__ out1, float* __restrict__ emax1,
                         float* __restrict__ den1, float* __restrict__ out2,
                         float* __restrict__ emax2, float* __restrict__ den2, int n) {
    int i = blockIdx.x * blockDim.x + threadIdx.x;
    if (i < n * HID) out1[i] = 0.0f;
    if (i < n * NC2) out2[i] = 0.0f;
    if (i < n * NH1) { emax1[i] = -INFINITY; den1[i] = 0.0f; }
    if (i < n)       { emax2[i] = -INFINITY; den2[i] = 0.0f; }
}

// ---------------- GEMM1: h1 = relu(x) @ W1  (fp32 WMMA 16x16x4) ----------------
// One wave computes a 16-row x 256-col panel: A frag loaded once per k-step,
// reused across all 16 column tiles (16 v8f accumulators).

__global__ __launch_bounds__(128, 1)
void dbp_gemm1(const float* __restrict__ x, const float* __restrict__ W1,
               float* __restrict__ h1, int n) {
    const int lane = threadIdx.x & 31;
    const int wave = (blockIdx.x * blockDim.x + threadIdx.x) >> 5;
    const int m0   = wave * 16;
    if (m0 >= n) return;                       // wave-uniform: EXEC stays all-ones for WMMA
    const int col  = lane & 15;
    const int koff = (lane < 16) ? 0 : 2;      // ISA A layout: lanes 0-15 K=0/1, 16-31 K=2/3
    int row = m0 + col; if (row >= n) row = n - 1;
    const float* xrow = x + (size_t)row * HID;

    v8f acc[16];
#pragma unroll
    for (int j = 0; j < 16; ++j) acc[j] = (v8f)(0.0f);

    for (int k = 0; k < HID; k += 4) {
        float a0 = fmaxf(xrow[k + koff],     0.0f);   // fused relu(x)
        float a1 = fmaxf(xrow[k + koff + 1], 0.0f);
        v2f a; a.x = a0; a.y = a1;
        const float* wk = W1 + (size_t)(k + koff) * HID + col;
#pragma unroll
        for (int j = 0; j < 16; ++j) {
            v2f b; b.x = wk[j * 16]; b.y = wk[j * 16 + HID];
            acc[j] = __builtin_amdgcn_wmma_f32_16x16x4_f32(
                false, a, false, b, (short)0, acc[j], false, false);
        }
    }

    const int rbase = m0 + ((lane >> 4) << 3); // C/D layout: lanes 16-31 hold M=8..15
#pragma unroll
    for (int j = 0; j < 16; ++j) {
#pragma unroll
        for (int r = 0; r < 8; ++r) {
            int rr = rbase + r;
            if (rr < n) h1[(size_t)rr * HID + j * 16 + col] = acc[j][r];
        }
    }
}

// ---------------- attention coefficients, layer 1 ----------------

__global__ void dbp_att1(const float* __restrict__ h1, const float* __restrict__ a1s,
                         const float* __restrict__ a1d, float* __restrict__ as,
                         float* __restrict__ ad, int n) {
    int t = blockIdx.x * blockDim.x + threadIdx.x;
    if (t >= n * NH1) return;
    int node = t >> 3, h = t & 7;
    const float* hp = h1 + (size_t)node * HID + h * NC1;
    const float* sv = a1s + h * NC1;
    const float* dv = a1d + h * NC1;
    float ss = 0.0f, dd = 0.0f;
#pragma unroll
    for (int c = 0; c < NC1; ++c) { float v = hp[c]; ss += v * sv[c]; dd += v * dv[c]; }
    as[t] = ss; ad[t] = dd;
}

// ---------------- edge passes, layer 1 ----------------

__global__ void dbp_emax1(const int* __restrict__ ei, int E, int ET,
                          const float* __restrict__ as, const float* __restrict__ ad,
                          float* __restrict__ emax) {
    int e = blockIdx.x * blockDim.x + threadIdx.x;
    if (e >= ET) return;
    int s, d; edge_src_dst(ei, E, e, s, d);
#pragma unroll
    for (int h = 0; h < NH1; ++h) {
        float el = as[s * NH1 + h] + ad[d * NH1 + h];
        el = (el > 0.0f) ? el : el * NEG_SLOPE;
        atomic_max_f32(&emax[d * NH1 + h], el);
    }
}

__global__ void dbp_esum1(const int* __restrict__ ei, int E, int ET,
                          const float* __restrict__ as, const float* __restrict__ ad,
                          const float* __restrict__ emax, float* __restrict__ ex1,
                          float* __restrict__ den) {
    int e = blockIdx.x * blockDim.x + threadIdx.x;
    if (e >= ET) return;
    int s, d; edge_src_dst(ei, E, e, s, d);
#pragma unroll
    for (int h = 0; h < NH1; ++h) {
        float el = as[s * NH1 + h] + ad[d * NH1 + h];
        el = (el > 0.0f) ? el : el * NEG_SLOPE;
        float ex = __expf(el - emax[d * NH1 + h]);
        ex1[(size_t)e * NH1 + h] = ex;
        atomicAdd(&den[d * NH1 + h], ex);
    }
}

// heavy pass: gather h1[src] (float4) * alpha, scatter-add into out1[dst] (L2 resident)
__global__ void dbp_eagg1(const int* __restrict__ ei, int E, int ET,
                          const float* __restrict__ h1, const float* __restrict__ ex1,
                          const float* __restrict__ den, float* __restrict__ out1) {
    int t = blockIdx.x * blockDim.x + threadIdx.x;
    if (t >= ET * 64) return;
    int e = t >> 6, q = t & 63;                 // q: 4-channel group, 64 groups = 256 ch
    int s, d; edge_src_dst(ei, E, e, s, d);
    int h = q >> 3;
    float alpha = ex1[(size_t)e * NH1 + h] / den[d * NH1 + h];
    const float4 hv = *reinterpret_cast<const float4*>(h1 + (size_t)s * HID + q * 4);
    float* o = out1 + (size_t)d * HID + q * 4;
    atomicAdd(o + 0, hv.x * alpha);
    atomicAdd(o + 1, hv.y * alpha);
    atomicAdd(o + 2, hv.z * alpha);
    atomicAdd(o + 3, hv.w * alpha);
}

// ---------------- GEMM2: h2 = relu(out1 + b1) @ W2  (fp32 WMMA, single 16-col tile) ----

__global__ __launch_bounds__(128, 1)
void dbp_gemm2(const float* __restrict__ out1, const float* __restrict__ b1,
               const float* __restrict__ W2, float* __restrict__ h2, int n) {
    const int lane = threadIdx.x & 31;
    const int wave = (blockIdx.x * blockDim.x + threadIdx.x) >> 5;
    const int m0   = wave * 16;
    if (m0 >= n) return;
    const int col  = lane & 15;
    const int koff = (lane < 16) ? 0 : 2;
    int row = m0 + col; if (row >= n) row = n - 1;
    const float* arow = out1 + (size_t)row * HID;

    v8f acc = (v8f)(0.0f);
    for (int k = 0; k < HID; k += 4) {
        float a0 = fmaxf(arow[k + koff]     + b1[k + koff],     0.0f); // fused +b1, relu
        float a1 = fmaxf(arow[k + koff + 1] + b1[k + koff + 1], 0.0f);
        v2f a; a.x = a0; a.y = a1;
        v2f b; b.x = W2[(size_t)(k + koff) * NC2 + col];
               b.y = W2[(size_t)(k + koff + 1) * NC2 + col];
        acc = __builtin_amdgcn_wmma_f32_16x16x4_f32(
            false, a, false, b, (short)0, acc, false, false);
    }

    const int rbase = m0 + ((lane >> 4) << 3);
#pragma unroll
    for (int r = 0; r < 8; ++r) {
        int rr = rbase + r;
        if (rr < n) h2[(size_t)rr * NC2 + col] = acc[r];
    }
}

// ---------------- attention + edge passes, layer 2 (1 head, 16 ch) ----------------

__global__ void dbp_att2(const float* __restrict__ h2, const float* __restrict__ a2s,
                         const float* __restrict__ a2d, float* __restrict__ as,
                         float* __restrict__ ad, int n) {
    int node = blockIdx.x * blockDim.x + threadIdx.x;
    if (node >= n) return;
    const float* hp = h2 + (size_t)node * NC2;
    float ss = 0.0f, dd = 0.0f;
#pragma unroll
    for (int c = 0; c < NC2; ++c) { float v = hp[c]; ss += v * a2s[c]; dd += v * a2d[c]; }
    as[node] = ss; ad[node] = dd;
}

__global__ void dbp_emax2(const int* __restrict__ ei, int E, int ET,
                          const float* __restrict__ as, const float* __restrict__ ad,
                          float* __restrict__ emax) {
    int e = blockIdx.x * blockDim.x + threadIdx.x;
    if (e >= ET) return;
    int s, d; edge_src_dst(ei, E, e, s, d);
    float el = as[s] + ad[d];
    el = (el > 0.0f) ? el : el * NEG_SLOPE;
    atomic_max_f32(&emax[d], el);
}

__global__ void dbp_esum2(const int* __restrict__ ei, int E, int ET,
                          const float* __restrict__ as, const float* __restrict__ ad,
                          const float* __restrict__ emax, float* __restrict__ ex2,
                          float* __restrict__ den) {
    int e = blockIdx.x * blockDim.x + threadIdx.x;
    if (e >= ET) return;
    int s, d; edge_src_dst(ei, E, e, s, d);
    float el = as[s] + ad[d];
    el = (el > 0.0f) ? el : el * NEG_SLOPE;
    float ex = __expf(el - emax[d]);
    ex2[e] = ex;
    atomicAdd(&den[d], ex);
}

__global__ void dbp_eagg2(const int* __restrict__ ei, int E, int ET,
                          const float* __restrict__ h2, const float* __restrict__ ex2,
                          const float* __restrict__ den, float* __restrict__ out2) {
    int t = blockIdx.x * blockDim.x + threadIdx.x;
    if (t >= ET * 4) return;
    int e = t >> 2, c4 = (t & 3) * 4;
    int s, d; edge_src_dst(ei, E, e, s, d);
    float alpha = ex2[e] / den[d];
    const float4 hv = *reinterpret_cast<const float4*>(h2 + (size_t)s * NC2 + c4);
    float* o = out2 + (size_t)d * NC2 + c4;
    atomicAdd(o + 0, hv.x * alpha);
    atomicAdd(o + 1, hv.y * alpha);
    atomicAdd(o + 2, hv.z * alpha);
    atomicAdd(o + 3, hv.w * alpha);
}

// ---------------- final: softmax(out2 + b2) over 16 channels ----------------

__global__ void dbp_final(const float* __restrict__ out2, const float* __restrict__ b2,
                          float* __restrict__ y, int n) {
    int node = blockIdx.x * blockDim.x + threadIdx.x;
    if (node >= n) return;
    const float* p = out2 + (size_t)node * NC2;
    float v[NC2];
    float m = -INFINITY;
#pragma unroll
    for (int c = 0; c < NC2; ++c) { v[c] = p[c] + b2[c]; m = fmaxf(m, v[c]); }
    float ssum = 0.0f;
#pragma unroll
    for (int c = 0; c < NC2; ++c) { v[c] = __expf(v[c] - m); ssum += v[c]; }
    float inv = 1.0f / ssum;
    float* yo = y + (size_t)node * NC2;
#pragma unroll
    for (int c = 0; c < NC2; ++c) yo[c] = v[c] * inv;
}

// ---------------- launch ----------------

static inline unsigned nblk(long long t, int b) { return (unsigned)((t + b - 1) / b); }

extern "C" void kernel_launch(void* const* d_in, const int* in_sizes, int n_in,
                              void* d_out, int out_size, void* d_ws, size_t ws_size,
                              hipStream_t stream) {
    (void)n_in; (void)out_size; (void)ws_size;
    const float* x   = (const float*)d_in[0];
    const int*   ei  = (const int*)d_in[1];
    const float* W1  = (const float*)d_in[2];
    const float* a1s = (const float*)d_in[3];
    const float* a1d = (const float*)d_in[4];
    const float* b1  = (const float*)d_in[5];
    const float* W2  = (const float*)d_in[6];
    const float* a2s = (const float*)d_in[7];
    const float* a2d = (const float*)d_in[8];
    const float* b2  = (const float*)d_in[9];
    float* y = (float*)d_out;

    const int n  = in_sizes[0] / HID;   // 50000
    const int E  = in_sizes[1] / 2;     // 800000
    const int ET = E + n;               // + self loops

    float* ws = (float*)d_ws;
    size_t o = 0;
    float* h1    = ws + o; o += (size_t)n * HID;
    float* out1  = ws + o; o += (size_t)n * HID;
    float* as1   = ws + o; o += (size_t)n * NH1;
    float* ad1   = ws + o; o += (size_t)n * NH1;
    float* emax1 = ws + o; o += (size_t)n * NH1;
    float* den1  = ws + o; o += (size_t)n * NH1;
    float* ex1   = ws + o; o += (size_t)ET * NH1;
    float* h2    = ws + o; o += (size_t)n * NC2;
    float* out2  = ws + o; o += (size_t)n * NC2;
    float* as2   = ws + o; o += (size_t)n;
    float* ad2   = ws + o; o += (size_t)n;
    float* emax2 = ws + o; o += (size_t)n;
    float* den2  = ws + o; o += (size_t)n;
    float* ex2   = ws + o; o += (size_t)ET;

    const int BT = 256;
    const long long waves = (n + 15) / 16;

    dbp_init <<<nblk((long long)n * HID, BT), BT, 0, stream>>>(out1, emax1, den1,
                                                               out2, emax2, den2, n);
    // layer 1
    dbp_gemm1<<<nblk(waves * 32, 128), 128, 0, stream>>>(x, W1, h1, n);
    dbp_att1 <<<nblk((long long)n * NH1, BT), BT, 0, stream>>>(h1, a1s, a1d, as1, ad1, n);
    dbp_emax1<<<nblk(ET, BT), BT, 0, stream>>>(ei, E, ET, as1, ad1, emax1);
    dbp_esum1<<<nblk(ET, BT), BT, 0, stream>>>(ei, E, ET, as1, ad1, emax1, ex1, den1);
    dbp_eagg1<<<nblk((long long)ET * 64, BT), BT, 0, stream>>>(ei, E, ET, h1, ex1, den1, out1);
    // layer 2
    dbp_gemm2<<<nblk(waves * 32, 128), 128, 0, stream>>>(out1, b1, W2, h2, n);
    dbp_att2 <<<nblk(n, BT), BT, 0, stream>>>(h2, a2s, a2d, as2, ad2, n);
    dbp_emax2<<<nblk(ET, BT), BT, 0, stream>>>(ei, E, ET, as2, ad2, emax2);
    dbp_esum2<<<nblk(ET, BT), BT, 0, stream>>>(ei, E, ET, as2, ad2, emax2, ex2, den2);
    dbp_eagg2<<<nblk((long long)ET * 4, BT), BT, 0, stream>>>(ei, E, ET, h2, ex2, den2, out2);

    dbp_final<<<nblk(n, BT), BT, 0, stream>>>(out2, b2, y, n);
}